// SchNETInteractionBlock_4904852652344
// MI455X (gfx1250) — compile-verified
//
#include <hip/hip_runtime.h>
#include <hip/hip_bf16.h>

typedef __attribute__((ext_vector_type(2))) float v2f;
typedef __attribute__((ext_vector_type(8))) float v8f;
typedef unsigned int u32x4 __attribute__((ext_vector_type(4)));
typedef int          i32x4 __attribute__((ext_vector_type(4)));
typedef int          i32x8 __attribute__((ext_vector_type(8)));

#define BN   32768   // B*N
#define DIM  128     // D
#define NF   64      // F
#define NR   20      // R
#define NP   1048576 // P

__device__ __forceinline__ float ssp_f(float v) {
    // shifted softplus: softplus(v) - ln(2), numerically stable
    float ax = fabsf(v);
    float sp = fmaxf(v, 0.0f) + log1pf(__expf(-ax));
    return sp - 0.69314718055994530942f;
}

// ---------------------------------------------------------------------------
// zero the atomic accumulator (must run every call: harness replays graph)
// ---------------------------------------------------------------------------
__global__ void k_zero(float4* __restrict__ p, int n4) {
    int i = blockIdx.x * blockDim.x + threadIdx.x;
    if (i < n4) p[i] = make_float4(0.f, 0.f, 0.f, 0.f);
}

// ---------------------------------------------------------------------------
// h = x @ W_in + b_in         [BN,128] x [128,64] -> [BN,64]
// block = 128 thr (4 waves); block covers 64 rows x 64 cols;
// wave w owns N-tile [16w,16w+16), loops 4 M-subtiles of 16.
// ---------------------------------------------------------------------------
__global__ __launch_bounds__(128) void k_h_gemm(
        const float* __restrict__ x, const float* __restrict__ W_in,
        const float* __restrict__ b_in, float* __restrict__ h) {
    const int lane = threadIdx.x & 31;
    const int wave = threadIdx.x >> 5;
    const int lo   = lane & 15;
    const int kb   = (lane < 16) ? 0 : 2;
    const int n    = wave * 16 + lo;          // B/C column for this lane
    const float bias = b_in[n];
    const int rowBlock = blockIdx.x * 64;

    for (int mt = 0; mt < 4; ++mt) {
        const int row0 = rowBlock + mt * 16;
        v8f c = {};
        const float* arow = x + (size_t)(row0 + lo) * DIM + kb;
        #pragma unroll
        for (int k = 0; k < DIM; k += 4) {
            v2f a; a.x = arow[k];                      a.y = arow[k + 1];
            v2f b; b.x = W_in[(size_t)(k + kb) * NF + n];
                   b.y = W_in[(size_t)(k + kb + 1) * NF + n];
            c = __builtin_amdgcn_wmma_f32_16x16x4_f32(
                    false, a, false, b, (short)0, c, false, false);
        }
        const int mbase = (lane < 16) ? 0 : 8;
        #pragma unroll
        for (int v = 0; v < 8; ++v) {
            const int m = v + mbase;
            h[(size_t)(row0 + m) * NF + n] = c[v] + bias;
        }
    }
}

// ---------------------------------------------------------------------------
// fused: Wij = ssp(f_ij @ W_f + b_f) * rcut ; x_ij = h[idx_j]*Wij ;
//        atomic scatter-add into agg[idx_i]
// block = 128 thr (4 waves); block covers 64 pairs x 64 filters.
// f_ij tile (64x20 f32 = 5120B contiguous) staged into LDS by the Tensor
// Data Mover (one tensor_load_to_lds per block); A-fragments then come from
// LDS (ds_load_b64). W_f slice register-resident per wave.
// ---------------------------------------------------------------------------
__global__ __launch_bounds__(128) void k_pair(
        const float* __restrict__ f_ij, const float* __restrict__ rcut,
        const float* __restrict__ W_f,  const float* __restrict__ b_f,
        const int* __restrict__ idx_i,  const int* __restrict__ idx_j,
        const float* __restrict__ h,    float* __restrict__ agg) {
    __shared__ float s_fij[64 * NR];   // 5120 B pair-feature tile
    __shared__ float s_rcut[64];
    __shared__ int   s_i[64], s_j[64];

    const int lane = threadIdx.x & 31;
    const int wave = threadIdx.x >> 5;
    const int lo   = lane & 15;
    const int kb   = (lane < 16) ? 0 : 2;
    const int n    = wave * 16 + lo;
    const float bias = b_f[n];
    const int pairBlock = blockIdx.x * 64;

    // keep this wave's W_f slice (20 x 16) in registers: 5 K-steps
    v2f breg[5];
    #pragma unroll
    for (int s = 0; s < 5; ++s) {
        const int k = s * 4 + kb;
        breg[s].x = W_f[(size_t)k * NF + n];
        breg[s].y = W_f[(size_t)(k + 1) * NF + n];
    }

    // stage per-pair scalars
    if (threadIdx.x < 64) {
        s_rcut[threadIdx.x] = rcut[pairBlock + threadIdx.x];
        s_i[threadIdx.x]    = idx_i[pairBlock + threadIdx.x];
        s_j[threadIdx.x]    = idx_j[pairBlock + threadIdx.x];
    }

    // --- TDM: DMA the contiguous f_ij tile into LDS (wave 0 only) ---------
    if (wave == 0) {
        const unsigned long long ga =
            (unsigned long long)(uintptr_t)(f_ij + (size_t)pairBlock * NR);
        const unsigned int lds_off =
            (unsigned int)(uintptr_t)&s_fij[0];   // LDS aperture: low 32b = offset
        // D# group0: count=1 | lds_addr | global_addr[56:0] | type=2
        u32x4 g0 = { 1u, lds_off, (unsigned int)ga,
                     (unsigned int)(ga >> 32) | (2u << 30) };
        // D# group1: data_size=4B(code2); tensor_dim0=1280; tensor_dim1=1;
        //            tile_dim0=1280 (1D tile); tensor_dim0_stride=1280
        i32x8 g1 = { (int)0x00020000,       // wg_mask=0, data_size=2
                     (int)(1280u << 16),    // tensor_dim0[15:0] << 16
                     (int)(1u << 16),       // tensor_dim0 hi=0 | tensor_dim1 lo=1
                     (int)(1280u << 16),    // tensor_dim1 hi=0 | tile_dim0=1280
                     0,                     // tile_dim1=0 (unused), tile_dim2=0
                     1280,                  // tensor_dim0_stride lo
                     0, 0 };
        i32x4 gz4 = { 0, 0, 0, 0 };
#if __has_include(<hip/amd_detail/amd_gfx1250_TDM.h>)
        i32x8 gz8 = { 0, 0, 0, 0, 0, 0, 0, 0 };
        __builtin_amdgcn_tensor_load_to_lds(g0, g1, gz4, gz4, gz8, 0);
#else
        __builtin_amdgcn_tensor_load_to_lds(g0, g1, gz4, gz4, 0);
#endif
        __builtin_amdgcn_s_wait_tensorcnt(0);
    }
    // warm L2 for the next block's tile (global_prefetch_b8)
    if (wave == 1 && pairBlock + 128 <= NP) {
        __builtin_prefetch(f_ij + (size_t)(pairBlock + 64) * NR, 0, 1);
    }
    __syncthreads();

    for (int pt = 0; pt < 4; ++pt) {
        v8f c = {};
        const float* arow = &s_fij[(pt * 16 + lo) * NR + kb];   // LDS reads
        #pragma unroll
        for (int s = 0; s < 5; ++s) {
            v2f a; a.x = arow[s * 4]; a.y = arow[s * 4 + 1];
            c = __builtin_amdgcn_wmma_f32_16x16x4_f32(
                    false, a, false, breg[s], (short)0, c, false, false);
        }

        const int mbase = (lane < 16) ? 0 : 8;
        #pragma unroll
        for (int v = 0; v < 8; ++v) {
            const int m = pt * 16 + v + mbase;
            const float w   = ssp_f(c[v] + bias) * s_rcut[m];
            const float xij = h[(size_t)s_j[m] * NF + n] * w;    // L2-resident gather
            unsafeAtomicAdd(&agg[(size_t)s_i[m] * NF + n], xij); // global_atomic_add_f32
        }
    }
}

// ---------------------------------------------------------------------------
// out = ssp(agg @ W_out + b_out)   [BN,64] x [64,128] -> [BN,128]
// block = 256 thr (8 waves); block covers 64 rows x 128 cols;
// wave w owns N-tile [16w,16w+16); W_out slice register-resident (32 VGPRs).
// ---------------------------------------------------------------------------
__global__ __launch_bounds__(256) void k_out(
        const float* __restrict__ agg, const float* __restrict__ W_out,
        const float* __restrict__ b_out, float* __restrict__ out) {
    const int lane = threadIdx.x & 31;
    const int wave = threadIdx.x >> 5;
    const int lo   = lane & 15;
    const int kb   = (lane < 16) ? 0 : 2;
    const int n    = wave * 16 + lo;
    const float bias = b_out[n];

    v2f breg[16];
    #pragma unroll
    for (int s = 0; s < 16; ++s) {
        const int k = s * 4 + kb;
        breg[s].x = W_out[(size_t)k * DIM + n];
        breg[s].y = W_out[(size_t)(k + 1) * DIM + n];
    }

    const int rowBlock = blockIdx.x * 64;
    for (int mt = 0; mt < 4; ++mt) {
        const int row0 = rowBlock + mt * 16;
        v8f c = {};
        const float* arow = agg + (size_t)(row0 + lo) * NF + kb;
        #pragma unroll
        for (int s = 0; s < 16; ++s) {
            v2f a; a.x = arow[s * 4]; a.y = arow[s * 4 + 1];
            c = __builtin_amdgcn_wmma_f32_16x16x4_f32(
                    false, a, false, breg[s], (short)0, c, false, false);
        }
        const int mbase = (lane < 16) ? 0 : 8;
        #pragma unroll
        for (int v = 0; v < 8; ++v) {
            const int m = v + mbase;
            out[(size_t)(row0 + m) * DIM + n] = ssp_f(c[v] + bias);
        }
    }
}

// ---------------------------------------------------------------------------
extern "C" void kernel_launch(void* const* d_in, const int* in_sizes, int n_in,
                              void* d_out, int out_size, void* d_ws, size_t ws_size,
                              hipStream_t stream) {
    const float* x     = (const float*)d_in[0];
    const float* f_ij  = (const float*)d_in[1];
    const float* rcut  = (const float*)d_in[2];
    const float* W_in  = (const float*)d_in[3];
    const float* b_in  = (const float*)d_in[4];
    const float* W_f   = (const float*)d_in[5];
    const float* b_f   = (const float*)d_in[6];
    const float* W_out = (const float*)d_in[7];
    const float* b_out = (const float*)d_in[8];
    const int*   idx_i = (const int*)d_in[9];
    const int*   idx_j = (const int*)d_in[10];
    float* out = (float*)d_out;

    float* h   = (float*)d_ws;                       // [BN, 64]  8 MB
    float* agg = h + (size_t)BN * NF;                // [BN, 64]  8 MB

    // zero accumulator (atomics), every call
    {
        const int n4 = (BN * NF) / 4;
        k_zero<<<(n4 + 255) / 256, 256, 0, stream>>>((float4*)agg, n4);
    }
    // h = x @ W_in + b_in
    k_h_gemm<<<BN / 64, 128, 0, stream>>>(x, W_in, b_in, h);
    // fused filter-net + gather + scatter-add (TDM-staged f_ij tiles)
    k_pair<<<NP / 64, 128, 0, stream>>>(f_ij, rcut, W_f, b_f, idx_i, idx_j, h, agg);
    // out = ssp(agg @ W_out + b_out)
    k_out<<<BN / 64, 256, 0, stream>>>(agg, W_out, b_out, out);
}